// CurvatureLoss_46557445488914
// MI455X (gfx1250) — compile-verified
//
#include <hip/hip_runtime.h>

typedef float v2f __attribute__((ext_vector_type(2)));
typedef float v8f __attribute__((ext_vector_type(8)));

#define NPTS  4096
#define BATCH 4
#define TILES (NPTS / 16)
#define DBL_BIG 1.0e300
#define EPS 1e-12f

// Single-instruction f64 min/max via inline CDNA5 asm. Bypasses LLVM's
// canonicalization and can never be turned into divergent control flow.
__device__ inline double dmin(double a, double b)
{
    double r;
    asm("v_min_num_f64 %0, %1, %2" : "=v"(r) : "v"(a), "v"(b));
    return r;
}
__device__ inline double dmax(double a, double b)
{
    double r;
    asm("v_max_num_f64 %0, %1, %2" : "=v"(r) : "v"(a), "v"(b));
    return r;
}

// Packed sort key: f64 value whose numeric order == score order, with the
// candidate index embedded in the low 12 mantissa bits (f32->f64 leaves the
// low 29 mantissa bits zero; idx < 4096 fits in 12). Built from 32-bit
// halves so only ONE v_or_b32 (on the low dword) is emitted per key.
__device__ inline double pack_key(float s, int idx)
{
    const double ds = (double)s;
    const int lo = __double2loint(ds) | idx;
    const int hi = __double2hiint(ds);
    return __hiloint2double(hi, lo);
}

__device__ inline int key_idx(double k)
{
    return __double2loint(k) & 0xFFF;
}

// Insert key into ascending sorted list kk[0..KEEP-1] using only f64 min/max:
// 2*(KEEP-1)+1 single-instruction ops, fully branch-free.
template<int KEEP>
__device__ inline void insert_key(double (&kk)[KEEP], double k)
{
    double cur = k;
#pragma unroll
    for (int j = 0; j < KEEP - 1; ++j) {
        const double mn = dmin(cur, kk[j]);
        const double mx = dmax(cur, kk[j]);
        kk[j] = mn;
        cur   = mx;
    }
    kk[KEEP - 1] = dmin(cur, kk[KEEP - 1]);
}

// One wave per 16-query tile. Computes score s(m,n) = -2*q_m . c_n + |c_n|^2
// = |c_n - q_m|^2 - |q_m|^2 (argmin-equivalent to squared distance) via
// V_WMMA_F32_16X16X4_F32 with homogeneous coords:
//   A row = (x,y,z,1), B col = (-2x,-2y,-2z,|c|^2).
//
// DROP_SELF (self-KNN): the self candidate is the strict row minimum, so we
// keep top-3 and drop the smallest after the merge — exactly the reference's
// top_k(k+1)[1:] semantics, with zero per-candidate exclusion logic.
template<int KEEP, bool DROP_SELF>
__global__ __launch_bounds__(32) void knn_wmma_kernel(
    const float* __restrict__ Qp,   // queries    [B][N][3]
    const float* __restrict__ Cp,   // candidates [B][N][3]
    int* __restrict__ out_idx)      // [B*N][KEEP - (DROP_SELF?1:0)]
{
    const int wid   = blockIdx.x;           // b*TILES + qtile
    const int b     = wid / TILES;
    const int qtile = wid - b * TILES;
    const int lane  = threadIdx.x;
    const int col   = lane & 15;
    const int hi    = lane >> 4;

    const float* Qb = Qp + (size_t)b * NPTS * 3;
    const float* Cb = Cp + (size_t)b * NPTS * 3;

    // A matrix (16x4 f32): lanes 0-15 hold K=0,1 ; lanes 16-31 hold K=2,3 ; M = lane%16
    const int qm = qtile * 16 + col;
    const float qx = Qb[3 * qm + 0];
    const float qy = Qb[3 * qm + 1];
    const float qz = Qb[3 * qm + 2];
    v2f A;
    A.x = hi ? qz : qx;     // K=0 (x) | K=2 (z)
    A.y = hi ? 1.0f : qy;   // K=1 (y) | K=3 (1)

    // running per-lane top-KEEP keys for the 8 query rows this lane holds in D
    double kk[8][KEEP];
#pragma unroll
    for (int r = 0; r < 8; ++r)
#pragma unroll
        for (int j = 0; j < KEEP; ++j) kk[r][j] = DBL_BIG;

#pragma unroll 2
    for (int t = 0; t < TILES; ++t) {
        const int cidx = t * 16 + col;
        const float cx = Cb[3 * cidx + 0];
        const float cy = Cb[3 * cidx + 1];
        const float cz = Cb[3 * cidx + 2];
        const float nb2 = cx * cx + cy * cy + cz * cz;

        // B matrix (4x16 f32): lanes 0-15 hold K=0,1 ; lanes 16-31 hold K=2,3 ; N = lane%16
        v2f Bv;
        Bv.x = hi ? (-2.0f * cz) : (-2.0f * cx);
        Bv.y = hi ? nb2 : (-2.0f * cy);

        v8f c = {0.f, 0.f, 0.f, 0.f, 0.f, 0.f, 0.f, 0.f};
        v8f d = __builtin_amdgcn_wmma_f32_16x16x4_f32(
            /*neg_a=*/false, A, /*neg_b=*/false, Bv,
            /*c_mod=*/(short)0, c, /*reuse_a=*/false, /*reuse_b=*/false);

        // D layout: lane%16 = candidate column N, VGPR slot r = query row (hi ? r+8 : r)
#pragma unroll
        for (int r = 0; r < 8; ++r) {
            insert_key<KEEP>(kk[r], pack_key(d[r], cidx));
        }
    }

    // merge the 16 lane-columns per query row through LDS
    __shared__ double lk[KEEP][256];
#pragma unroll
    for (int r = 0; r < 8; ++r) {
        const int m = hi ? (r + 8) : r;
#pragma unroll
        for (int j = 0; j < KEEP; ++j) lk[j][m * 16 + col] = kk[r][j];
    }
    __syncthreads();

    if (lane < 16) {
        double best[KEEP];
#pragma unroll
        for (int j = 0; j < KEEP; ++j) best[j] = DBL_BIG;
        for (int cc = 0; cc < 16; ++cc) {
#pragma unroll
            for (int j = 0; j < KEEP; ++j) {
                insert_key<KEEP>(best, lk[j][lane * 16 + cc]);
            }
        }
        const int q = b * NPTS + qtile * 16 + lane;
        if (DROP_SELF) {
            // best[0] is the self match (strict row minimum); emit ranks 1..KEEP-1
            out_idx[(size_t)q * (KEEP - 1) + 0] = key_idx(best[1]);
            if (KEEP > 2) out_idx[(size_t)q * (KEEP - 1) + 1] = key_idx(best[2]);
        } else {
            out_idx[(size_t)q] = key_idx(best[0]);
        }
    }
}

__device__ inline float kappa2(const float* __restrict__ Pb, int n, int j1, int j2,
                               float nx, float ny, float nz)
{
    const float px = Pb[3 * n + 0], py = Pb[3 * n + 1], pz = Pb[3 * n + 2];
    float acc = 0.0f;
    const int jj[2] = { j1, j2 };
#pragma unroll
    for (int k = 0; k < 2; ++k) {
        const int j = jj[k];
        float vx = Pb[3 * j + 0] - px;
        float vy = Pb[3 * j + 1] - py;
        float vz = Pb[3 * j + 2] - pz;
        const float inv = 1.0f / sqrtf(vx * vx + vy * vy + vz * vz + EPS);
        acc += fabsf((vx * nx + vy * ny + vz * nz) * inv);
    }
    return 0.5f * acc;
}

__global__ __launch_bounds__(256) void loss_kernel(
    const float* __restrict__ ori, const float* __restrict__ adv,
    const float* __restrict__ nrm,
    const int* __restrict__ nbr_ori, const int* __restrict__ nbr_adv,
    const int* __restrict__ nn_idx, float* __restrict__ partials)
{
    const int gid = blockIdx.x * 256 + threadIdx.x;   // 0 .. B*N-1
    const int b   = gid / NPTS;
    const int n   = gid - b * NPTS;
    const float* Ob = ori + (size_t)b * NPTS * 3;
    const float* Ab = adv + (size_t)b * NPTS * 3;
    const float* Nb = nrm + (size_t)b * NPTS * 3;

    // ori kappa with ori normal
    const float onx = Nb[3 * n + 0], ony = Nb[3 * n + 1], onz = Nb[3 * n + 2];
    const float ok = kappa2(Ob, n, nbr_ori[2 * gid + 0], nbr_ori[2 * gid + 1], onx, ony, onz);

    // adv kappa with normal gathered from nearest ori point
    const int ni = nn_idx[gid];
    const float anx = Nb[3 * ni + 0], any_ = Nb[3 * ni + 1], anz = Nb[3 * ni + 2];
    const float ak = kappa2(Ab, n, nbr_adv[2 * gid + 0], nbr_adv[2 * gid + 1], anx, any_, anz);

    const float diff = ak - ok;

    __shared__ float red[256];
    red[threadIdx.x] = diff * diff;
    __syncthreads();
#pragma unroll
    for (int s = 128; s > 0; s >>= 1) {
        if (threadIdx.x < s) red[threadIdx.x] += red[threadIdx.x + s];
        __syncthreads();
    }
    if (threadIdx.x == 0) partials[blockIdx.x] = red[0];
}

__global__ __launch_bounds__(64) void final_kernel(
    const float* __restrict__ partials, float* __restrict__ out, int nblocks)
{
    __shared__ float red[64];
    red[threadIdx.x] = (threadIdx.x < nblocks) ? partials[threadIdx.x] : 0.0f;
    __syncthreads();
#pragma unroll
    for (int s = 32; s > 0; s >>= 1) {
        if (threadIdx.x < s) red[threadIdx.x] += red[threadIdx.x + s];
        __syncthreads();
    }
    if (threadIdx.x == 0) out[0] = red[0] / (float)(BATCH * NPTS);
}

extern "C" void kernel_launch(void* const* d_in, const int* in_sizes, int n_in,
                              void* d_out, int out_size, void* d_ws, size_t ws_size,
                              hipStream_t stream)
{
    const float* ori = (const float*)d_in[0];
    const float* adv = (const float*)d_in[1];
    const float* nrm = (const float*)d_in[2];

    char* ws = (char*)d_ws;
    int*   nbr_ori  = (int*)ws;                                            // B*N*2 ints
    int*   nbr_adv  = (int*)(ws + (size_t)BATCH * NPTS * 2 * sizeof(int)); // B*N*2 ints
    int*   nn_idx   = (int*)(ws + (size_t)BATCH * NPTS * 4 * sizeof(int)); // B*N ints
    float* partials = (float*)(ws + (size_t)BATCH * NPTS * 5 * sizeof(int)); // 64 floats

    const dim3 gknn(BATCH * TILES);
    const dim3 bknn(32);
    knn_wmma_kernel<3, true ><<<gknn, bknn, 0, stream>>>(ori, ori, nbr_ori);
    knn_wmma_kernel<3, true ><<<gknn, bknn, 0, stream>>>(adv, adv, nbr_adv);
    knn_wmma_kernel<1, false><<<gknn, bknn, 0, stream>>>(adv, ori, nn_idx);

    const int nblocks = (BATCH * NPTS) / 256;  // 64
    loss_kernel<<<nblocks, 256, 0, stream>>>(ori, adv, nrm, nbr_ori, nbr_adv, nn_idx, partials);
    final_kernel<<<1, 64, 0, stream>>>(partials, (float*)d_out, nblocks);
}